// Domain_Model_11596411699935
// MI455X (gfx1250) — compile-verified
//
#include <hip/hip_runtime.h>

typedef __attribute__((ext_vector_type(2))) float v2f;
typedef __attribute__((ext_vector_type(4))) float v4f;
typedef __attribute__((ext_vector_type(8))) float v8f;

#define S_  16
#define P_  65536
#define N_  256
#define K_  32

// ---------------------------------------------------------------------------
// Kernel 1: zero-fill the 201 MB output with non-temporal b128 stores.
// This is >99.9% of all memory traffic; everything else is noise.
// ---------------------------------------------------------------------------
__global__ __launch_bounds__(256)
void zero_fill_kernel(v4f* __restrict__ out, long n4) {
  long i = (long)blockIdx.x * blockDim.x + threadIdx.x;
  const long stride = (long)gridDim.x * blockDim.x;
  v4f z = {0.0f, 0.0f, 0.0f, 0.0f};
  for (; i < n4; i += stride) {
    __builtin_nontemporal_store(z, &out[i]);
  }
}

// ---------------------------------------------------------------------------
// Kernel 2: one wave32 per row n. Lane k owns indices[n][k].
//  - duplicate count m via 32 intra-wave shuffles (self-inclusive, m>=1)
//  - gather params[s][p][0..3] as one b128
//  - projection v = PROJ @ params^T via V_WMMA_F32_16X16X4_F32:
//      A (16x4) = PROJ zero-padded: rows 0..2 = [[0,0,1,1],[0,1,0,0],[0,0,0,1]]
//      B (4x16) = params of 16 pairs (two passes for the 32 pairs)
//      D (16x16): lane j<16 holds channel c for pair j in VGPR c (c=0,1,2)
//  - store 1 - (1-v)^m to the three output planes (duplicates write
//    identical values -> deterministic)
// ---------------------------------------------------------------------------
__global__ __launch_bounds__(32)
void scatter_kernel(const float* __restrict__ schema_params,
                    const int* __restrict__ schema_ids,
                    const int* __restrict__ indices,
                    float* __restrict__ out) {
  const int n    = blockIdx.x;
  const int lane = threadIdx.x;

  const int s = schema_ids[n];
  const int p = indices[n * K_ + lane];

  // m = multiplicity of this lane's index within the row
  int m = 0;
#pragma unroll
  for (int j = 0; j < 32; ++j) {
    m += (__shfl(p, j, 32) == p) ? 1 : 0;
  }

  // gather the 4 contiguous schema params for (s, p): 16B-aligned b128
  const v4f q = *(const v4f*)(schema_params + ((size_t)s * P_ + (size_t)p) * 4);

  // A = PROJ padded to 16x4. ISA 32-bit A layout (16x4 MxK):
  //   lanes 0-15 : row M=lane,   VGPR0 = K0, VGPR1 = K1
  //   lanes16-31 : row M=lane-16, VGPR0 = K2, VGPR1 = K3
  // Nonzero entries: PROJ[0][2], PROJ[0][3], PROJ[1][1], PROJ[2][3]
  v2f a;
  a[0] = (lane == 16) ? 1.0f : 0.0f;
  a[1] = (lane == 1 || lane == 16 || lane == 18) ? 1.0f : 0.0f;

  float* __restrict__ out0 = out;
  float* __restrict__ out1 = out + (size_t)N_ * P_;
  float* __restrict__ out2 = out + 2 * (size_t)N_ * P_;

#pragma unroll
  for (int half = 0; half < 2; ++half) {
    const int src = half * 16 + (lane & 15);   // which pair this column serves

    // B (4x16 KxN), K-striped like A: lanes 0-15 -> K0/K1, lanes 16-31 -> K2/K3
    const float g0 = __shfl(q.x, src, 32);
    const float g1 = __shfl(q.y, src, 32);
    const float g2 = __shfl(q.z, src, 32);
    const float g3 = __shfl(q.w, src, 32);
    v2f b;
    b[0] = (lane < 16) ? g0 : g2;
    b[1] = (lane < 16) ? g1 : g3;

    v8f c = {};
    // 8 args: (neg_a, A, neg_b, B, c_mod, C, reuse_a, reuse_b)
    v8f d = __builtin_amdgcn_wmma_f32_16x16x4_f32(
        false, a, false, b, (short)0, c, false, false);

    const int pp = __shfl(p, src, 32);
    const int mm = __shfl(m, src, 32);

    if (lane < 16) {
      // D layout: VGPR r, lanes 0-15 -> (M=r, N=lane). M = channel, N = pair.
      const size_t o = (size_t)n * P_ + (size_t)pp;
      const float b0 = 1.0f - d[0];
      const float b1 = 1.0f - d[1];
      const float b2 = 1.0f - d[2];
      float r0 = b0, r1 = b1, r2 = b2;
      for (int t = 1; t < mm; ++t) { r0 *= b0; r1 *= b1; r2 *= b2; }
      out0[o] = 1.0f - r0;
      out1[o] = 1.0f - r1;
      out2[o] = 1.0f - r2;
    }
  }
}

extern "C" void kernel_launch(void* const* d_in, const int* in_sizes, int n_in,
                              void* d_out, int out_size, void* d_ws, size_t ws_size,
                              hipStream_t stream) {
  (void)in_sizes; (void)n_in; (void)d_ws; (void)ws_size;
  const float* schema_params = (const float*)d_in[0];
  const int*   schema_ids    = (const int*)d_in[1];
  const int*   indices       = (const int*)d_in[2];
  float*       out           = (float*)d_out;

  // out_size = 3 * N * P = 50,331,648 floats -> 12,582,912 float4
  const long n4 = (long)out_size / 4;
  const int  zb = 12288;  // 12288 * 256 threads * 4 iters covers n4 exactly
  zero_fill_kernel<<<zb, 256, 0, stream>>>((v4f*)out, n4);

  scatter_kernel<<<N_, 32, 0, stream>>>(schema_params, schema_ids, indices, out);
}